// DoubleChannelColorLoss_71442486001672
// MI455X (gfx1250) — compile-verified
//
#include <hip/hip_runtime.h>

typedef float v4f __attribute__((ext_vector_type(4)));
typedef float v2f __attribute__((ext_vector_type(2)));
typedef float v8f __attribute__((ext_vector_type(8)));

namespace {
constexpr int kB    = 32;
constexpr int kC    = 3;
constexpr int kH    = 512;
constexpr int kW    = 512;
constexpr int kHW4  = kH * kW / 4;   // channel stride in float4 units (65536)
constexpr int kRow4 = kW / 4;        // row stride in float4 units (128)
constexpr int kBlkX = kW / 64;       // 8 blocks across W (64 columns each)
constexpr int kNPart = kBlkX * kB;   // 256 partial sums
}

__device__ __forceinline__ float qclamp(float x) {
  // floor(clip(x, 0, 255)) — applied once per column (commutes with min/max)
  return floorf(fminf(fmaxf(x, 0.0f), 255.0f));
}

// Stage 1: streaming column min/max of the V channel + per-block loss partial.
// grid = (kW/64, kB), block = 256 threads (8 wave32s).
__global__ __launch_bounds__(256) void dc_stage1(
    const float* __restrict__ pred, const float* __restrict__ tgt,
    float* __restrict__ partial) {
  __shared__ v4f s_pmin[256], s_pmax[256], s_tmin[256], s_tmax[256];  // 16 KB
  __shared__ float s_loss[64];

  const int tid  = threadIdx.x;
  const int g    = tid & 15;                 // float4 column group (16/block)
  const int s    = tid >> 4;                 // 32-row H segment (16/column)
  const int b    = blockIdx.y;
  const int col4 = blockIdx.x * 16 + g;      // float4 column index in [0,128)

  const v4f* __restrict__ p4 = (const v4f*)pred + (size_t)b * (kC * kHW4) + col4;
  const v4f* __restrict__ t4 = (const v4f*)tgt  + (size_t)b * (kC * kHW4) + col4;

  v4f pmin = { 3.0e38f,  3.0e38f,  3.0e38f,  3.0e38f};
  v4f pmax = {-3.0e38f, -3.0e38f, -3.0e38f, -3.0e38f};
  v4f tmin = pmin, tmax = pmax;

  const int h0 = s * 32;
  #pragma unroll 4
  for (int r = 0; r < 32; ++r) {
    const int off = (h0 + r) * kRow4;

    // global_prefetch_b8, 8 rows ahead on all six streams (speculative => OOB-safe)
    __builtin_prefetch(p4 + off + 8 * kRow4);
    __builtin_prefetch(p4 + kHW4 + off + 8 * kRow4);
    __builtin_prefetch(p4 + 2 * kHW4 + off + 8 * kRow4);
    __builtin_prefetch(t4 + off + 8 * kRow4);
    __builtin_prefetch(t4 + kHW4 + off + 8 * kRow4);
    __builtin_prefetch(t4 + 2 * kHW4 + off + 8 * kRow4);

    // single-use streaming data: non-temporal b128 loads
    v4f p0 = __builtin_nontemporal_load(p4 + off);
    v4f p1 = __builtin_nontemporal_load(p4 + kHW4 + off);
    v4f p2 = __builtin_nontemporal_load(p4 + 2 * kHW4 + off);
    v4f vm = __builtin_elementwise_max(__builtin_elementwise_max(p0, p1), p2);
    pmin = __builtin_elementwise_min(pmin, vm);
    pmax = __builtin_elementwise_max(pmax, vm);

    v4f q0 = __builtin_nontemporal_load(t4 + off);
    v4f q1 = __builtin_nontemporal_load(t4 + kHW4 + off);
    v4f q2 = __builtin_nontemporal_load(t4 + 2 * kHW4 + off);
    v4f wm = __builtin_elementwise_max(__builtin_elementwise_max(q0, q1), q2);
    tmin = __builtin_elementwise_min(tmin, wm);
    tmax = __builtin_elementwise_max(tmax, wm);
  }

  s_pmin[tid] = pmin; s_pmax[tid] = pmax;
  s_tmin[tid] = tmin; s_tmax[tid] = tmax;
  __syncthreads();

  // combine the 16 H-segments for each of the 64 columns, quantize, square-diff
  if (tid < 64) {
    const int gg = tid >> 2, j = tid & 3;
    float mnp = 3.0e38f, mxp = -3.0e38f, mnt = 3.0e38f, mxt = -3.0e38f;
    #pragma unroll
    for (int ss = 0; ss < 16; ++ss) {
      const int idx = ss * 16 + gg;
      mnp = fminf(mnp, s_pmin[idx][j]);
      mxp = fmaxf(mxp, s_pmax[idx][j]);
      mnt = fminf(mnt, s_tmin[idx][j]);
      mxt = fmaxf(mxt, s_tmax[idx][j]);
    }
    const float dd = qclamp(mnp) - qclamp(mnt);  // dark term
    const float db = qclamp(mxp) - qclamp(mxt);  // bright term
    s_loss[tid] = dd * dd + db * db;
  }
  __syncthreads();

  if (tid == 0) {
    float acc = 0.0f;
    #pragma unroll
    for (int i = 0; i < 64; ++i) acc += s_loss[i];
    partial[blockIdx.y * gridDim.x + blockIdx.x] = acc;
  }
}

// Stage 2: one fully-active wave32 reduces the 256 partials.
// Final 32->1 collapse rides the matrix engine: A = all-ones 16x4 f32 so
// D[m][n] = sum_k B[k][n]; lane partials fill B's first VGPR (second = 0),
// so summing D row 0 over the wave yields 2x the grand total (rows duplicate).
__global__ __launch_bounds__(32) void dc_stage2(
    const float* __restrict__ partial, float* __restrict__ out) {
  const int lane = threadIdx.x;  // exactly 32 threads => EXEC all ones for WMMA
  float s = 0.0f;
  #pragma unroll
  for (int i = 0; i < kNPart / 32; ++i) s += partial[lane + i * 32];

  v2f a;  a[0] = 1.0f; a[1] = 1.0f;   // A: 16x4 all-ones
  v2f bm; bm[0] = s;   bm[1] = 0.0f;  // B: 4x16, one lane value per slot
  v8f c = {};
  c = __builtin_amdgcn_wmma_f32_16x16x4_f32(false, a, false, bm, (short)0, c,
                                            false, false);
  float t = c[0];
  #pragma unroll
  for (int off = 16; off > 0; off >>= 1) t += __shfl_xor(t, off, 32);
  if (lane == 0) out[0] = (t * 0.5f) * (1.0f / 16384.0f);  // /= B*W (dark+bright share it)
}

extern "C" void kernel_launch(void* const* d_in, const int* in_sizes, int n_in,
                              void* d_out, int out_size, void* d_ws, size_t ws_size,
                              hipStream_t stream) {
  const float* pred = (const float*)d_in[0];
  const float* tgt  = (const float*)d_in[1];
  float* out     = (float*)d_out;
  float* partial = (float*)d_ws;   // 256 floats of scratch

  dim3 grid(kBlkX, kB);            // (8, 32) = 256 blocks, 8 waves each
  dc_stage1<<<grid, 256, 0, stream>>>(pred, tgt, partial);
  dc_stage2<<<1, 32, 0, stream>>>(partial, out);
}